// ClusterNormZCA_37151467110664
// MI455X (gfx1250) — compile-verified
//
#include <hip/hip_runtime.h>

// ZCA whitening (ClusterNorm) for x[256][64][4096] f32 on gfx1250 (MI455X).
// One workgroup (256 thr = 8 wave32) per batch. All matmuls via
// V_WMMA_F32_16X16X4_F32; tile staging via GLOBAL_LOAD_ASYNC_TO_LDS_B128
// (double-buffered, ASYNCcnt-synchronized).
//
// Math (identical to reference):
//   cov  = X*X^T/M - mu*mu^T            (Gram trick: no per-element centering)
//   covS = rho*(tr/p)*I + (1-rho)*cov   (Ledoit-Wolf)
//   S    = covS^(-1/2)                  (coupled Newton-Schulz, 16 iters)
//   Z    = S*X - (S*mu)*1^T

typedef __attribute__((ext_vector_type(2))) float v2f;
typedef __attribute__((ext_vector_type(8))) float v8f;
typedef __attribute__((ext_vector_type(4))) int   v4i;

#define BATCH   256
#define CDIM    64
#define MDIM    4096
#define TILE_M  256
#define NTILES  (MDIM / TILE_M)        // 16
#define XS_STRIDE 260                  // 256 + 4: 16B-aligned rows, 64-bank spread
#define XS_SZ   (CDIM * XS_STRIDE)     // 16640 floats
#define NS_STRIDE 66                   // 64 + 2 pad
#define NS_SZ   (64 * NS_STRIDE)       // 4224 floats
#define NS_ITERS 16

#if __has_builtin(__builtin_amdgcn_global_load_async_to_lds_b128)
#define USE_ASYNC 1
// Builtin prototype (from hipcc diagnostic): param0 = v4i in addrspace(1)
// (printed as "__device__"), param1 = LDS side, then offset, cpol.
#define ASYNC_B128(gp, lp)                                                   \
  __builtin_amdgcn_global_load_async_to_lds_b128(                            \
      (__attribute__((address_space(1))) v4i*)(uintptr_t)(gp),               \
      (__attribute__((address_space(3))) v4i*)(uint32_t)(uintptr_t)(lp),     \
      0, 0)
#else
#define USE_ASYNC 0
#endif

#if __has_builtin(__builtin_amdgcn_s_wait_asynccnt)
#define WAIT_ASYNC(n) __builtin_amdgcn_s_wait_asynccnt(n)
#else
#define WAIT_ASYNC(n) asm volatile("s_wait_asynccnt %0" ::"i"(n) : "memory")
#endif

__device__ __forceinline__ v8f wmma_f32(v2f a, v2f b, v8f c) {
  return __builtin_amdgcn_wmma_f32_16x16x4_f32(
      false, a, false, b, (short)0, c, false, false);
}

// Stage one raw 64x256 f32 tile of x into LDS (padded stride). 8 waves, each
// issues 16 x b128. Async path: no VGPR bounce, tracked by ASYNCcnt.
__device__ __forceinline__ void stage_tile(const float* __restrict__ xb,
                                           float* __restrict__ dst,
                                           int m0, int wave, int lane) {
#if USE_ASYNC
  #pragma unroll
  for (int i = 0; i < 16; ++i) {
    const int idx = (i * 8 + wave) * 32 + lane;   // float4 index in tile
    const int c = idx >> 6, f4 = idx & 63;
    const float* gp = xb + c * MDIM + m0 + f4 * 4;
    float* lp = dst + c * XS_STRIDE + f4 * 4;     // 16B aligned
    ASYNC_B128(gp, lp);
  }
#else
  #pragma unroll 4
  for (int i = 0; i < 16; ++i) {
    const int idx = (i * 8 + wave) * 32 + lane;
    const int c = idx >> 6, f4 = idx & 63;
    float4 v = *(const float4*)(xb + c * MDIM + m0 + f4 * 4);
    *(float4*)(dst + c * XS_STRIDE + f4 * 4) = v;
  }
#endif
}

// 64x64 * 64x64 -> 64x64 LDS matmul (stride NS_STRIDE), 8 waves, no barrier.
// D = alpha*(A x B) + beta*I
__device__ __forceinline__ void mm64(float* __restrict__ D,
                                     const float* __restrict__ A,
                                     const float* __restrict__ Bm,
                                     float alpha, float beta,
                                     int wave, int lane) {
  const int half = lane >> 4, r = lane & 15;
  const int i = wave >> 1;
  #pragma unroll
  for (int jj = 0; jj < 2; ++jj) {
    const int j = (wave & 1) * 2 + jj;
    v8f acc = {};
    #pragma unroll 4
    for (int k = 0; k < 64; k += 4) {
      v2f a = *(const v2f*)&A[(i * 16 + r) * NS_STRIDE + k + 2 * half];
      v2f b;
      b.x = Bm[(k + 2 * half)     * NS_STRIDE + j * 16 + r];
      b.y = Bm[(k + 2 * half + 1) * NS_STRIDE + j * 16 + r];
      acc = wmma_f32(a, b, acc);
    }
    #pragma unroll
    for (int v = 0; v < 8; ++v) {
      const int row = i * 16 + v + 8 * half;
      const int col = j * 16 + r;
      float val = alpha * acc[v];
      if (beta != 0.0f && row == col) val += beta;
      D[row * NS_STRIDE + col] = val;
    }
  }
}

__global__ __launch_bounds__(256)
void zca_whiten_kernel(const float* __restrict__ x, float* __restrict__ out) {
  __shared__ float pool[2 * XS_SZ + NS_SZ];   // 37504 f (~147 KB of 320 KB WGP LDS)
  __shared__ float mu[CDIM], smu[CDIM];
  __shared__ float s_tr, s_tr2;

  const int b    = blockIdx.x;
  const int tid  = threadIdx.x;
  const int lane = tid & 31;
  const int wave = tid >> 5;
  const int half = lane >> 4;
  const int r    = lane & 15;
  const float* xb = x   + (size_t)b * CDIM * MDIM;
  float*       ob = out + (size_t)b * CDIM * MDIM;

  float* xs0 = pool;
  float* xs1 = pool + XS_SZ;
  float* S   = pool + 2 * XS_SZ;     // S_hat lives past the tile buffers

  // ---------- Phase 1: row means ----------
  {
    const int c = tid >> 2, q = tid & 3;
    const float* p = xb + c * MDIM + q * (MDIM / 4);
    float s = 0.f;
    for (int i = 0; i < MDIM / 4; i += 4) {
      float4 v = *(const float4*)(p + i);
      s += v.x + v.y + v.z + v.w;
    }
    pool[tid] = s;
  }
  __syncthreads();
  if (tid < CDIM)
    mu[tid] = (pool[4 * tid] + pool[4 * tid + 1] +
               pool[4 * tid + 2] + pool[4 * tid + 3]) * (1.0f / MDIM);
  if (tid == 0) { s_tr = 0.f; s_tr2 = 0.f; }
  __syncthreads();

  // ---------- Phase 2: Gram G = X*X^T via WMMA, async double-buffered ----------
  const int bi  = wave >> 1;           // cov block row (0..3)
  const int bj0 = (wave & 1) * 2;      // cov block cols bj0, bj0+1
  v8f acc0 = {}, acc1 = {};
  stage_tile(xb, xs0, 0, wave, lane);
  for (int t = 0; t < NTILES; ++t) {
    if (t + 1 < NTILES) {
      stage_tile(xb, (t & 1) ? xs0 : xs1, (t + 1) * TILE_M, wave, lane);
#if USE_ASYNC
      WAIT_ASYNC(16);                  // tile t landed (own), barrier => all waves
#endif
    } else {
#if USE_ASYNC
      WAIT_ASYNC(0);
#endif
    }
    __syncthreads();
    const float* xs = (t & 1) ? xs1 : xs0;
    #pragma unroll 4
    for (int k = 0; k < TILE_M; k += 4) {
      v2f a  = *(const v2f*)&xs[(bi * 16 + r) * XS_STRIDE + k + 2 * half];
      v2f b0 = *(const v2f*)&xs[(bj0 * 16 + r) * XS_STRIDE + k + 2 * half];
      v2f b1 = *(const v2f*)&xs[((bj0 + 1) * 16 + r) * XS_STRIDE + k + 2 * half];
      acc0 = wmma_f32(a, b0, acc0);
      acc1 = wmma_f32(a, b1, acc1);
    }
    __syncthreads();
  }

  float* Y = pool;                     // NS buffers overlay tile region (synced)
  float* Z = pool + NS_SZ;
  float* T = pool + 2 * NS_SZ;
  float* U = pool + 3 * NS_SZ;

  // cov = G/M - mu*mu^T
  #pragma unroll
  for (int v = 0; v < 8; ++v) {
    const int row = bi * 16 + v + 8 * half;
    const int c0 = bj0 * 16 + r, c1 = c0 + 16;
    Y[row * NS_STRIDE + c0] = acc0[v] * (1.0f / MDIM) - mu[row] * mu[c0];
    Y[row * NS_STRIDE + c1] = acc1[v] * (1.0f / MDIM) - mu[row] * mu[c1];
  }
  __syncthreads();

  // ---------- Phase 3: Ledoit-Wolf shrinkage + trace normalization ----------
  {
    float l_tr = 0.f, l_tr2 = 0.f;
    #pragma unroll
    for (int k = 0; k < 16; ++k) {
      const int e = k * 256 + tid;
      const int i = e >> 6, j = e & 63;
      const float v = Y[i * NS_STRIDE + j];
      l_tr2 += v * v;                  // C symmetric: tr(C*C) = ||C||_F^2
      if (i == j) l_tr += v;
    }
    atomicAdd(&s_tr, l_tr);            // ds_add_f32
    atomicAdd(&s_tr2, l_tr2);
  }
  __syncthreads();
  const float tr = s_tr, t2 = s_tr2;
  const float nf = (float)MDIM, pf = (float)CDIM;
  const float num = (nf - 2.0f) / nf * t2 + tr * tr;
  const float den = (nf + 2.0f) * (t2 - tr * tr / pf);
  const float rho = fminf(num / den, 1.0f);
  const float itr = 1.0f / tr;         // Y0 eigenvalues land in (0,1]
  #pragma unroll
  for (int k = 0; k < 16; ++k) {
    const int e = k * 256 + tid;
    const int i = e >> 6, j = e & 63;
    const float v = Y[i * NS_STRIDE + j];
    const float sh = (1.0f - rho) * v + ((i == j) ? rho * (tr / pf) : 0.0f);
    Y[i * NS_STRIDE + j] = sh * itr;
    Z[i * NS_STRIDE + j] = (i == j) ? 1.0f : 0.0f;
  }
  __syncthreads();

  // ---------- Phase 4: coupled Newton-Schulz -> Z = (C/tr)^(-1/2) ----------
  for (int it = 0; it < NS_ITERS; ++it) {
    mm64(T, Z, Y, -0.5f, 1.5f, wave, lane);   // T = 1.5I - 0.5*Z*Y
    __syncthreads();
    mm64(U, Y, T, 1.0f, 0.0f, wave, lane);    // Ynew = Y*T
    __syncthreads();
    mm64(Y, T, Z, 1.0f, 0.0f, wave, lane);    // Znew = T*Z (old Y slot)
    __syncthreads();
    float* oy = Y; Y = U; U = Z; Z = oy;
  }

  // S_hat = Z / sqrt(tr)  (copied out of the tile-buffer overlay region)
  {
    const float fs = rsqrtf(tr);
    #pragma unroll
    for (int k = 0; k < 16; ++k) {
      const int e = k * 256 + tid;
      const int i = e >> 6, j = e & 63;
      S[i * NS_STRIDE + j] = Z[i * NS_STRIDE + j] * fs;
    }
  }
  __syncthreads();
  if (tid < CDIM) {                    // smu = S_hat * mu
    float s = 0.f;
    for (int k = 0; k < CDIM; ++k) s += S[tid * NS_STRIDE + k] * mu[k];
    smu[tid] = s;
  }
  __syncthreads();

  // ---------- Phase 5: out = S_hat*X - smu, async double-buffered ----------
  stage_tile(xb, xs0, 0, wave, lane);
  for (int t = 0; t < NTILES; ++t) {
    if (t + 1 < NTILES) {
      stage_tile(xb, (t & 1) ? xs0 : xs1, (t + 1) * TILE_M, wave, lane);
#if USE_ASYNC
      WAIT_ASYNC(16);
#endif
    } else {
#if USE_ASYNC
      WAIT_ASYNC(0);
#endif
    }
    __syncthreads();
    const float* xs = (t & 1) ? xs1 : xs0;
    const int m0 = t * TILE_M;
    for (int i = 0; i < 4; ++i) {
      v2f af[16];                      // S_hat row-panel i in registers
      #pragma unroll
      for (int k = 0; k < 16; ++k)
        af[k] = *(const v2f*)&S[(i * 16 + r) * NS_STRIDE + 4 * k + 2 * half];
      #pragma unroll
      for (int jj = 0; jj < 2; ++jj) {
        const int ml = (wave * 2 + jj) * 16;
        v8f acc = {};
        #pragma unroll
        for (int k = 0; k < 16; ++k) {
          const int k4 = 4 * k + 2 * half;
          v2f bf;
          bf.x = xs[k4 * XS_STRIDE + ml + r];
          bf.y = xs[(k4 + 1) * XS_STRIDE + ml + r];
          acc = wmma_f32(af[k], bf, acc);
        }
        #pragma unroll
        for (int v = 0; v < 8; ++v) {
          const int row = i * 16 + v + 8 * half;
          ob[(size_t)row * MDIM + m0 + ml + r] = acc[v] - smu[row];
        }
      }
    }
    __syncthreads();
  }
}

extern "C" void kernel_launch(void* const* d_in, const int* in_sizes, int n_in,
                              void* d_out, int out_size, void* d_ws, size_t ws_size,
                              hipStream_t stream) {
  const float* xp = (const float*)d_in[0];
  float* op = (float*)d_out;
  zca_whiten_kernel<<<BATCH, 256, 0, stream>>>(xp, op);
}